// VarReg_6253472383024
// MI455X (gfx1250) — compile-verified
//
#include <hip/hip_runtime.h>
#include <cstddef>

// ---------------------------------------------------------------------------
// Variational (demons) registration, 128^3 f32, 20 iterations.
// Gaussian smoothing passes are band-matrix GEMMs on V_WMMA_F32_16X16X4_F32.
// Weight fragments are precomputed once into a 3KB table (hot in WGP$/L2) and
// fetched with coalesced b64 loads; inner loops are pure load + wmma.
// ---------------------------------------------------------------------------

typedef __attribute__((ext_vector_type(2))) float v2f;
typedef __attribute__((ext_vector_type(8))) float v8f;

#define NVOX (128*128*128)   // 2,097,152 voxels
#define SD   (128*128)       // d stride
#define SH   128             // h stride

// Normalized Gaussian, sigma=1, radius=3 (used only in the init kernel).
__device__ __forceinline__ float gk(int i) {
    unsigned u = (unsigned)i;
    float r = 0.0f;
    r = (u == 0u || u == 6u) ? 0.004433048f : r;
    r = (u == 1u || u == 5u) ? 0.054005582f : r;
    r = (u == 2u || u == 4u) ? 0.242036223f : r;
    r = (u == 3u)            ? 0.399050293f : r;
    return r;
}

__device__ __forceinline__ int clamp127(int v) {
    return v < 0 ? 0 : (v > 127 ? 127 : v);
}

// jnp.gradient semantics along one axis (pos p in [0,128), given stride)
__device__ __forceinline__ float cgrad(const float* __restrict__ f, int idx,
                                       int p, int stride) {
    if (p == 0)   return f[idx + stride] - f[idx];
    if (p == 127) return f[idx] - f[idx - stride];
    return 0.5f * (f[idx + stride] - f[idx - stride]);
}

// ---------------- weight-fragment tables (one 192-thread block, runs once) --
// tw[t*32+lane]: W-pass B-fragment; tl[t*32+lane]: H/D-pass A-fragment.
__global__ __launch_bounds__(192) void wtab_k(v2f* __restrict__ tw,
                                              v2f* __restrict__ tl) {
    int idx = threadIdx.x;                 // 0..191 = t*32 + lane
    int t = idx >> 5, lane = idx & 31;
    int i = lane & 15, hi = lane >> 4;
    v2f w, l;
    w.x = gk(i - 2 * hi + 7 - 4 * t);      // B(k0+p, c0+j), (c0-k0)=4-4t
    w.y = gk(i - 2 * hi + 6 - 4 * t);
    l.x = gk(4 * t + 2 * hi - i - 1);      // A(m0+j, k) = g[4t+p-j-1]
    l.y = gk(4 * t + 2 * hi - i);
    tw[idx] = w;
    tl[idx] = l;
}

// ---------------- fixed-image gradient (3 channels) ------------------------
__global__ __launch_bounds__(256) void grad3_k(const float* __restrict__ f,
                                               float* __restrict__ g) {
    int idx = blockIdx.x * 256 + threadIdx.x;          // grid covers NVOX exactly
    int w = idx & 127, h = (idx >> 7) & 127, d = idx >> 14;
    g[idx]            = cgrad(f, idx, d, SD);
    g[NVOX + idx]     = cgrad(f, idx, h, SH);
    g[2 * NVOX + idx] = cgrad(f, idx, w, 1);
}

// ---------------- trilinear warp (map_coordinates order=1, mode='nearest') --
__global__ __launch_bounds__(256) void warp_k(const float* __restrict__ mov,
                                              const float* __restrict__ vf,
                                              float* __restrict__ wout) {
    int idx = blockIdx.x * 256 + threadIdx.x;
    int w = idx & 127, h = (idx >> 7) & 127, d = idx >> 14;
    float cd = (float)d + vf[idx];
    float ch = (float)h + vf[NVOX + idx];
    float cw = (float)w + vf[2 * NVOX + idx];
    cd = fminf(fmaxf(cd, -1.0f), 128.0f);
    ch = fminf(fmaxf(ch, -1.0f), 128.0f);
    cw = fminf(fmaxf(cw, -1.0f), 128.0f);
    float fd = floorf(cd), fh = floorf(ch), fw = floorf(cw);
    float td = cd - fd, th = ch - fh, tw = cw - fw;
    int d0 = clamp127((int)fd), d1 = clamp127((int)fd + 1);
    int h0 = clamp127((int)fh), h1 = clamp127((int)fh + 1);
    int w0 = clamp127((int)fw), w1 = clamp127((int)fw + 1);
    const float* p0 = mov + (size_t)d0 * SD;
    const float* p1 = mov + (size_t)d1 * SD;
    float v000 = p0[h0 * SH + w0], v001 = p0[h0 * SH + w1];
    float v010 = p0[h1 * SH + w0], v011 = p0[h1 * SH + w1];
    float v100 = p1[h0 * SH + w0], v101 = p1[h0 * SH + w1];
    float v110 = p1[h1 * SH + w0], v111 = p1[h1 * SH + w1];
    float a00 = v000 + tw * (v001 - v000);
    float a01 = v010 + tw * (v011 - v010);
    float a10 = v100 + tw * (v101 - v100);
    float a11 = v110 + tw * (v111 - v110);
    float b0 = a00 + th * (a01 - a00);
    float b1 = a10 + th * (a11 - a10);
    wout[idx] = b0 + td * (b1 - b0);
}

// ---------------- demon forces + Euler update u = vf + tau*f ---------------
__global__ __launch_bounds__(256) void forces_k(const float* __restrict__ wp,
                                                const float* __restrict__ fx,
                                                const float* __restrict__ fg,
                                                const float* __restrict__ vf,
                                                float* __restrict__ u) {
    int idx = blockIdx.x * 256 + threadIdx.x;
    int w = idx & 127, h = (idx >> 7) & 127, d = idx >> 14;
    float diff = wp[idx] - fx[idx];
    float g0 = 0.5f * (cgrad(wp, idx, d, SD) + fg[idx]);
    float g1 = 0.5f * (cgrad(wp, idx, h, SH) + fg[NVOX + idx]);
    float g2 = 0.5f * (cgrad(wp, idx, w, 1)  + fg[2 * NVOX + idx]);
    float denom = g0 * g0 + g1 * g1 + g2 * g2 + diff * diff;
    float s = (denom > 1e-6f) ? (diff / denom) : 0.0f;   // TAU = 1
    u[idx]            = vf[idx]            + s * g0;
    u[NVOX + idx]     = vf[NVOX + idx]     + s * g1;
    u[2 * NVOX + idx] = vf[2 * NVOX + idx] + s * g2;
}

// ---------------- W-axis blur: Out = In x Band, via WMMA f32 16x16x4 -------
// Rows = flattened (c,d,h) = 49152, cols = W = 128. One block = 16-row slab.
// LDS tile zero-padded by 4 guard columns each side -> A-fragment is one
// aligned ds_load_b64 per K-step, no masking. B-fragments come from the
// precomputed table (6 coalesced b64 loads per wave).
#define TPITCH 140   // stride mod 64 = 12, gcd(12,64)=4 -> 16 lanes, 16 banks
__global__ __launch_bounds__(256) void smooth_w_k(const float* __restrict__ in,
                                                  float* __restrict__ out,
                                                  const v2f* __restrict__ tw) {
    __shared__ float tile[16 * TPITCH];
    const int r0  = blockIdx.x * 16;
    const int tid = threadIdx.x;
    const int lane = tid & 31;
    // weight fragments: issue loads first, they overlap the staging below
    v2f bw[6];
#pragma unroll
    for (int t = 0; t < 6; ++t) bw[t] = tw[t * 32 + lane];
    // zero the guard columns: cols 0..3 and 132..135 of each row
    if (tid < 128) {
        int r = tid >> 3, pc = tid & 7;
        tile[r * TPITCH + (pc < 4 ? pc : 128 + pc)] = 0.0f;
    }
    // stage 16x128 slab: b128 global loads + b128 LDS stores (16B aligned)
    const float4* in4 = (const float4*)(in + (size_t)r0 * 128);
#pragma unroll
    for (int g = tid; g < 512; g += 256) {
        float4 v = in4[g];
        int r = g >> 5, c = (g & 31) * 4;
        *(float4*)&tile[r * TPITCH + 4 + c] = v;
    }
    __syncthreads();
    const int wave = tid >> 5;               // 8 waves -> 8 column tiles
    const int c0 = wave * 16;
    const int i  = lane & 15;                // A row / B,D column within tile
    const int hi = lane >> 4;                // K sub-group: p base = 2*hi
    v8f acc = {};
#pragma unroll
    for (int t = 0; t < 6; ++t) {
        int ka0 = c0 - 4 + 4 * t + 2 * hi;   // even; in [-4, 130]
        v2f a = *(const v2f*)&tile[i * TPITCH + 4 + ka0];  // guard cols = 0
        acc = __builtin_amdgcn_wmma_f32_16x16x4_f32(
            false, a, false, bw[t], (short)0, acc, false, false);
    }
#pragma unroll
    for (int v = 0; v < 8; ++v)
        out[(size_t)(r0 + v + 8 * hi) * 128 + (c0 + i)] = acc[v];
}

// ---------------- H/D-axis blur: Out = Band x In, via WMMA f32 16x16x4 -----
// out(m,n) = sum_k g[k-m+3] * in(k,n); strideK = row stride of k/m dimension.
// A-fragments from the precomputed table (valid for every m-tile); the block
// loops over all 8 m-tiles. K-windows are 4-aligned, so boundary steps are
// entirely out of range: skip t=0 at mt=0 and t=5 at mt=7 (zero contribution)
// with wave-uniform branches. All remaining loads are in-range and coalesced.
__global__ __launch_bounds__(256) void smooth_l_k(const float* __restrict__ in,
                                                  float* __restrict__ out,
                                                  int strideK, size_t sliceStride,
                                                  const v2f* __restrict__ tl) {
    const size_t base = (size_t)blockIdx.z * sliceStride;
    const int tid  = threadIdx.x;
    const int wave = tid >> 5, lane = tid & 31;
    const int n0 = blockIdx.x * 128 + wave * 16;
    const int j  = lane & 15;
    const int hi = lane >> 4;
    const float* inb = in + base;
    v2f wgt[6];
#pragma unroll
    for (int t = 0; t < 6; ++t) wgt[t] = tl[t * 32 + lane];
#pragma unroll 1
    for (int mt = 0; mt < 8; ++mt) {
        const int m0 = mt * 16;
        v8f acc = {};
#pragma unroll
        for (int t = 0; t < 6; ++t) {
            if (t == 0 && mt == 0) continue;   // k in [-4,-1]: zero-padded
            if (t == 5 && mt == 7) continue;   // k in [128,131]: zero-padded
            const int kb0 = m0 - 4 + 4 * t + 2 * hi;   // in [0,126]
            v2f b;
            b.x = inb[(size_t)kb0 * strideK + n0 + j];       // 16-lane contig
            b.y = inb[(size_t)(kb0 + 1) * strideK + n0 + j];
            acc = __builtin_amdgcn_wmma_f32_16x16x4_f32(
                false, wgt[t], false, b, (short)0, acc, false, false);
        }
#pragma unroll
        for (int v = 0; v < 8; ++v)
            out[base + (size_t)(m0 + v + 8 * hi) * strideK + n0 + j] = acc[v];
    }
}

// ---------------------------------------------------------------------------
extern "C" void kernel_launch(void* const* d_in, const int* in_sizes, int n_in,
                              void* d_out, int out_size, void* d_ws, size_t ws_size,
                              hipStream_t stream) {
    (void)in_sizes; (void)n_in; (void)out_size; (void)ws_size;
    const float* mov = (const float*)d_in[0];
    const float* fix = (const float*)d_in[1];
    float* vf = (float*)d_out;                        // (3, D, H, W) velocity

    float* wsf = (float*)d_ws;
    v2f* tw = (v2f*)wsf;                              // 192 v2f = 384 floats
    v2f* tl = (v2f*)(wsf + 384);                      // 192 v2f = 384 floats
    float* fg     = wsf + 1024;                       // 3*NVOX  fixed gradient
    float* bufA   = fg   + 3 * (size_t)NVOX;          // 3*NVOX  u / ping
    float* bufB   = bufA + 3 * (size_t)NVOX;          // 3*NVOX  pong
    float* warped = bufB;                             // NVOX (lifetime-disjoint)

    hipMemsetAsync(d_out, 0, 3 * (size_t)NVOX * sizeof(float), stream);

    dim3 blk(256);
    dim3 gv(NVOX / 256);                              // 8192 blocks, exact
    wtab_k<<<1, 192, 0, stream>>>(tw, tl);
    grad3_k<<<gv, blk, 0, stream>>>(fix, fg);

    for (int it = 0; it < 20; ++it) {
        warp_k  <<<gv, blk, 0, stream>>>(mov, vf, warped);
        forces_k<<<gv, blk, 0, stream>>>(warped, fix, fg, vf, bufA);
        // W pass: 3*128*128 rows / 16 = 3072 slabs
        smooth_w_k<<<dim3(3 * SD / 16), blk, 0, stream>>>(bufA, bufB, tw);
        // H pass: per (channel,d) slice; block covers all 8 m-tiles
        smooth_l_k<<<dim3(1, 1, 3 * 128), blk, 0, stream>>>(bufB, bufA, SH, (size_t)SD, tl);
        // D pass: per channel, n = flattened h*w; block covers all 8 m-tiles
        smooth_l_k<<<dim3(SD / 128, 1, 3), blk, 0, stream>>>(bufA, vf, SD, (size_t)NVOX, tl);
    }
}